// BatchedBlochWave_28887950032903
// MI455X (gfx1250) — compile-verified
//
#include <hip/hip_runtime.h>
#include <math.h>

// Bloch plane-wave synthesis as an fp32 complex GEMM on CDNA5 WMMA.
// psi[row, r] = (1/sqrt(vol)) * exp(i k_row . r) * sum_G cg[row,G] exp(i G.r)
// A: cg [128 x 8000] (re,im separate), B: phase [8000 G x 8000 r] synthesized
// in LDS from factorized per-axis tables exp(i m_d * u_d(r)).

typedef __attribute__((ext_vector_type(2))) float v2f;
typedef __attribute__((ext_vector_type(8))) float v8f;

#define N1D     20
#define NGRID   8000          // 20^3  (= nG = nR)
#define NROWS   128           // 2 spin * 4 k * 16 bands
#define KCHUNK  160           // K staging chunk (multiple of 4)
#define NCHUNK  (NGRID / KCHUNK)   // 50
#define LPAD    (KCHUNK + 4)  // padded LDS row stride (words) -> conflict-free
#define TWO_PI_F 6.28318530717958647692f

__global__ __launch_bounds__(256) void bloch_wmma_kernel(
    const float* __restrict__ Amat,   // [3][3]  lattice rows
    const float* __restrict__ kgrid,  // [4][3]
    const float* __restrict__ cg_re,  // [128][8000]
    const float* __restrict__ cg_im,  // [128][8000]
    const float* __restrict__ rpts,   // [8000][3]
    float* __restrict__ out)          // [128][8000][2] interleaved complex
{
    __shared__ float u_s[16][3];            // u = 2*pi * inv(A)^T r  per point
    __shared__ float r_s[16][3];            // raw r per point (for k-phase)
    __shared__ float tab[3][16][N1D][2];    // exp(i * f(idx) * u_dim) complex
    __shared__ float cos_s[16][LPAD];       // staged Re(phase) [pt][kk]
    __shared__ float sin_s[16][LPAD];       // staged Im(phase) [pt][kk]

    const int tid   = threadIdx.x;
    const int lane  = tid & 31;
    const int wave  = tid >> 5;             // 0..7 : row tile (one (s,k) pair)
    const int nbase = blockIdx.x * 16;      // this block's 16 r-points
    const int pt    = lane & 15;            // N index within tile
    const int koff  = (lane >> 4) * 2;      // K sub-offset per lane half

    // ---- 3x3 cofactors / det of A (cheap, every thread) ----
    const float a00=Amat[0],a01=Amat[1],a02=Amat[2];
    const float a10=Amat[3],a11=Amat[4],a12=Amat[5];
    const float a20=Amat[6],a21=Amat[7],a22=Amat[8];
    const float c00 =  (a11*a22 - a12*a21);
    const float c01 = -(a10*a22 - a12*a20);
    const float c02 =  (a10*a21 - a11*a20);
    const float c10 = -(a01*a22 - a02*a21);
    const float c11 =  (a00*a22 - a02*a20);
    const float c12 = -(a00*a21 - a01*a20);
    const float c20 =  (a01*a12 - a02*a11);
    const float c21 = -(a00*a12 - a02*a10);
    const float c22 =  (a00*a11 - a01*a10);
    const float det = a00*c00 + a01*c01 + a02*c02;

    // ---- per-point u vectors:  u_e = 2*pi/det * sum_d cof[e*?]...  ----
    // inv(A)[d][e] = cof[e][d]/det  =>  u_e = 2*pi/det * (c_e0*r0 + c_e1*r1 + c_e2*r2)
    if (tid < 16) {
        const float* rp = rpts + (size_t)(nbase + tid) * 3;
        const float r0 = rp[0], r1 = rp[1], r2 = rp[2];
        r_s[tid][0] = r0; r_s[tid][1] = r1; r_s[tid][2] = r2;
        const float s = TWO_PI_F / det;
        u_s[tid][0] = s * (c00*r0 + c01*r1 + c02*r2);
        u_s[tid][1] = s * (c10*r0 + c11*r1 + c12*r2);
        u_s[tid][2] = s * (c20*r0 + c21*r1 + c22*r2);
    }
    __syncthreads();

    // ---- factor tables: tab[dim][pt][idx] = exp(i * f(idx) * u[pt][dim]) ----
    for (int t = tid; t < 3 * 16 * N1D; t += 256) {
        const int dim = t / (16 * N1D);
        const int rem = t - dim * (16 * N1D);
        const int p   = rem / N1D;
        const int idx = rem - p * N1D;
        const float f   = (float)(idx < N1D / 2 ? idx : idx - N1D); // fftfreq*n
        const float ang = f * u_s[p][dim];
        float sn, cs;
        __sincosf(ang, &sn, &cs);
        tab[dim][p][idx][0] = cs;
        tab[dim][p][idx][1] = sn;
    }
    __syncthreads();

    // ---- accumulators: 4 real GEMM partials of the complex product ----
    v8f acc_rr = {0.f,0.f,0.f,0.f,0.f,0.f,0.f,0.f}; // Re(A)*cos
    v8f acc_ii = acc_rr;                            // Im(A)*sin
    v8f acc_ri = acc_rr;                            // Re(A)*sin
    v8f acc_ir = acc_rr;                            // Im(A)*cos

    const int mrow = wave * 16 + pt;                // this lane's A row
    const float* are_base = cg_re + (size_t)mrow * NGRID + koff;
    const float* aim_base = cg_im + (size_t)mrow * NGRID + koff;

    for (int ch = 0; ch < NCHUNK; ++ch) {
        const int k0 = ch * KCHUNK;

        // -- stage KCHUNK x 16 phases into LDS (2560 complex, 10/thread) --
        for (int t = tid; t < 16 * KCHUNK; t += 256) {
            const int kk = t >> 4;            // 0..KCHUNK-1
            const int p  = t & 15;            // point
            const int n  = k0 + kk;           // flat G index = i1*400+i2*20+i3
            const int i1 = n / 400;
            const int rm = n - i1 * 400;
            const int i2 = rm / 20;
            const int i3 = rm - i2 * 20;
            const float c1 = tab[0][p][i1][0], s1 = tab[0][p][i1][1];
            const float c2 = tab[1][p][i2][0], s2 = tab[1][p][i2][1];
            const float c3 = tab[2][p][i3][0], s3 = tab[2][p][i3][1];
            const float cr = c1 * c2 - s1 * s2;
            const float ci = c1 * s2 + s1 * c2;
            cos_s[p][kk] = cr * c3 - ci * s3;
            sin_s[p][kk] = cr * s3 + ci * c3;
        }
        __syncthreads();

        const float* are = are_base + k0;
        const float* aim = aim_base + k0;
        if (ch + 1 < NCHUNK) {                 // global_prefetch_b8 next A chunk
            __builtin_prefetch(are + KCHUNK, 0, 0);
            __builtin_prefetch(aim + KCHUNK, 0, 0);
        }

        // -- WMMA inner loop: 40 steps x 4 v_wmma_f32_16x16x4_f32 --
        #pragma unroll 4
        for (int kk = 0; kk < KCHUNK; kk += 4) {
            const v2f a_re = *(const v2f*)(are + kk);              // K=kk+koff..+1
            const v2f a_im = *(const v2f*)(aim + kk);
            const v2f b_c  = *(const v2f*)(&cos_s[pt][kk + koff]);
            const v2f b_s  = *(const v2f*)(&sin_s[pt][kk + koff]);
            acc_rr = __builtin_amdgcn_wmma_f32_16x16x4_f32(
                false, a_re, false, b_c, (short)0, acc_rr, false, false);
            acc_ii = __builtin_amdgcn_wmma_f32_16x16x4_f32(
                false, a_im, false, b_s, (short)0, acc_ii, false, false);
            acc_ri = __builtin_amdgcn_wmma_f32_16x16x4_f32(
                false, a_re, false, b_s, (short)0, acc_ri, false, false);
            acc_ir = __builtin_amdgcn_wmma_f32_16x16x4_f32(
                false, a_im, false, b_c, (short)0, acc_ir, false, false);
        }
        __syncthreads();
    }

    // ---- epilogue: psi = (S1-S2) + i(S3+S4), rotate by exp(i k.r)/sqrt(vol) ----
    const float* kv = kgrid + (size_t)(wave & 3) * 3;  // tile -> (s = wave>>2, k = wave&3)
    const float kr = kv[0]*r_s[pt][0] + kv[1]*r_s[pt][1] + kv[2]*r_s[pt][2];
    float sk, ck;
    __sincosf(kr, &sk, &ck);
    const float invsv = rsqrtf(fabsf(det));
    ck *= invsv;
    sk *= invsv;

    // C/D layout: lanes 0-15 -> M = j, lanes 16-31 -> M = j+8; N = lane&15
    const int row0 = wave * 16 + (lane >> 4) * 8;
    float* outp = out + ((size_t)row0 * NGRID + (nbase + pt)) * 2;
    #pragma unroll
    for (int j = 0; j < 8; ++j) {
        const float pre = acc_rr[j] - acc_ii[j];
        const float pim = acc_ri[j] + acc_ir[j];
        v2f o;
        o.x = pre * ck - pim * sk;
        o.y = pre * sk + pim * ck;
        *(v2f*)(outp + (size_t)j * NGRID * 2) = o;
    }
}

extern "C" void kernel_launch(void* const* d_in, const int* in_sizes, int n_in,
                              void* d_out, int out_size, void* d_ws, size_t ws_size,
                              hipStream_t stream) {
    const float* A     = (const float*)d_in[0];
    const float* kgrid = (const float*)d_in[1];
    const float* cg_re = (const float*)d_in[2];
    const float* cg_im = (const float*)d_in[3];
    const float* rpts  = (const float*)d_in[4];
    float* out = (float*)d_out;

    dim3 grid(NGRID / 16);   // 500 N-tiles
    dim3 block(256);         // 8 waves: one 16-row M-tile each
    hipLaunchKernelGGL(bloch_wmma_kernel, grid, block, 0, stream,
                       A, kgrid, cg_re, cg_im, rpts, out);
}